// ConditionalAutoencoderCC_59777354826304
// MI455X (gfx1250) — compile-verified
//
#include <hip/hip_runtime.h>

typedef __attribute__((ext_vector_type(2))) float v2f;
typedef __attribute__((ext_vector_type(8))) float v8f;

#define T_   128
#define N_   4000
#define P_   128
#define H0_  64
#define H1_  32
#define K_   8
#define BN_EPS 1e-5f

// ---------------------------------------------------------------------------
// Layer 0: H0pre[t,n,j] = xs[t,n,:] . W0[j,:] + b0[j]
// One wave = 16 rows x 64 cols (4 WMMA tiles), K=128 in steps of 4.
// ---------------------------------------------------------------------------
__global__ __launch_bounds__(256)
void gemm1_kernel(const float* __restrict__ xs, const float* __restrict__ W0,
                  const float* __restrict__ b0, float* __restrict__ H0pre) {
    __shared__ float W0T[P_ * H0_];            // [p][j], 32 KB
    const int tid = threadIdx.x;
    for (int i = tid; i < P_ * H0_; i += 256) {
        int j = i / P_, p = i % P_;            // W0 row-major read, coalesced
        W0T[p * H0_ + j] = W0[i];
    }
    __syncthreads();

    const int wave = tid >> 5, lane = tid & 31;
    const int hh = lane >> 4, l16 = lane & 15;
    const int mtile = blockIdx.x * 8 + wave;
    const int t = blockIdx.y;
    if (mtile >= N_ / 16) return;
    const int n0 = mtile * 16;

    const float* zrow = xs + ((size_t)t * N_ + (n0 + l16)) * P_;
    v8f acc[4] = {};
    for (int kk = 0; kk < P_; kk += 4) {
        const int k0 = kk + 2 * hh;
        v2f a;  a.x = zrow[k0];  a.y = zrow[k0 + 1];
#pragma unroll
        for (int j = 0; j < 4; ++j) {
            v2f b;
            b.x = W0T[k0 * H0_ + j * 16 + l16];
            b.y = W0T[(k0 + 1) * H0_ + j * 16 + l16];
            acc[j] = __builtin_amdgcn_wmma_f32_16x16x4_f32(
                false, a, false, b, (short)0, acc[j], false, false);
        }
    }
#pragma unroll
    for (int j = 0; j < 4; ++j) {
        const float bias = b0[j * 16 + l16];
#pragma unroll
        for (int i = 0; i < 8; ++i) {
            const int m = i + 8 * hh;
            H0pre[((size_t)t * N_ + n0 + m) * H0_ + j * 16 + l16] = acc[j][i] + bias;
        }
    }
}

// ---------------------------------------------------------------------------
// BatchNorm training stats over stock axis -> fused scale/shift per (t,f).
// ---------------------------------------------------------------------------
template <int F, int SL>
__global__ __launch_bounds__(256)
void bnstats_kernel(const float* __restrict__ H, const float* __restrict__ g,
                    const float* __restrict__ be, float* __restrict__ sc,
                    float* __restrict__ sh) {
    const int t = blockIdx.x, tid = threadIdx.x;
    const int f = tid % F, slice = tid / F;
    float s = 0.f, q = 0.f;
    for (int n = slice; n < N_; n += SL) {
        const float v = H[((size_t)t * N_ + n) * F + f];
        s += v; q += v * v;
    }
    __shared__ float rs[256], rq[256];
    rs[tid] = s; rq[tid] = q;
    __syncthreads();
    if (slice == 0) {
        for (int i = 1; i < SL; ++i) { s += rs[i * F + f]; q += rq[i * F + f]; }
        const float mu  = s * (1.f / N_);
        const float var = q * (1.f / N_) - mu * mu;   // biased, as torch/jnp
        const float r   = rsqrtf(var + BN_EPS);
        const float scale = g[f] * r;
        sc[t * F + f] = scale;
        sh[t * F + f] = be[f] - mu * scale;
    }
}

// ---------------------------------------------------------------------------
// Layer 1: H1pre = relu(bn(H0pre)) @ W1^T + b1 ; BN+ReLU fused into A-load.
// ---------------------------------------------------------------------------
__global__ __launch_bounds__(256)
void gemm2_kernel(const float* __restrict__ H0pre, const float* __restrict__ sc0,
                  const float* __restrict__ sh0, const float* __restrict__ W1,
                  const float* __restrict__ b1, float* __restrict__ H1pre) {
    __shared__ float W1T[H0_ * H1_];           // [k][j]
    __shared__ float s_sc[H0_], s_sh[H0_];
    const int tid = threadIdx.x;
    const int t = blockIdx.y;
    for (int i = tid; i < H0_ * H1_; i += 256) {
        int j = i / H0_, k = i % H0_;
        W1T[k * H1_ + j] = W1[i];
    }
    if (tid < H0_) { s_sc[tid] = sc0[t * H0_ + tid]; s_sh[tid] = sh0[t * H0_ + tid]; }
    __syncthreads();

    const int wave = tid >> 5, lane = tid & 31;
    const int hh = lane >> 4, l16 = lane & 15;
    const int mtile = blockIdx.x * 8 + wave;
    if (mtile >= N_ / 16) return;
    const int n0 = mtile * 16;

    const float* hrow = H0pre + ((size_t)t * N_ + (n0 + l16)) * H0_;
    v8f acc[2] = {};
    for (int kk = 0; kk < H0_; kk += 4) {
        const int k0 = kk + 2 * hh;
        v2f a;
        a.x = fmaxf(hrow[k0]     * s_sc[k0]     + s_sh[k0],     0.f);
        a.y = fmaxf(hrow[k0 + 1] * s_sc[k0 + 1] + s_sh[k0 + 1], 0.f);
#pragma unroll
        for (int j = 0; j < 2; ++j) {
            v2f b;
            b.x = W1T[k0 * H1_ + j * 16 + l16];
            b.y = W1T[(k0 + 1) * H1_ + j * 16 + l16];
            acc[j] = __builtin_amdgcn_wmma_f32_16x16x4_f32(
                false, a, false, b, (short)0, acc[j], false, false);
        }
    }
#pragma unroll
    for (int j = 0; j < 2; ++j) {
        const float bias = b1[j * 16 + l16];
#pragma unroll
        for (int i = 0; i < 8; ++i) {
            const int m = i + 8 * hh;
            H1pre[((size_t)t * N_ + n0 + m) * H1_ + j * 16 + l16] = acc[j][i] + bias;
        }
    }
}

// ---------------------------------------------------------------------------
// Gram: G[t] = xs[t]^T xs[t]  (128x128, K=4000).
// Block = one 16-row p strip; 8 waves cover the 8 q-tiles.
// xs panels (40 rows x 128 cols) are staged into LDS with the CDNA5 async
// copy path (global_load_async_to_lds_b128 / ASYNCcnt), double-buffered so
// stage s+1 HBM->LDS overlaps stage s WMMA. Rows padded to 132 floats.
// ---------------------------------------------------------------------------
#define NB_   40                  // panel rows per stage
#define NSTG_ (N_ / NB_)          // 100 stages
#define PSTR_ 132                 // padded LDS row stride in floats (528B, 16B-aligned)
#define CHNK_ (NB_ * P_ / 4 / 256)  // 5 x b128 chunks per thread per stage

__device__ __forceinline__ void gram_stage_issue(const float* __restrict__ g,
                                                 unsigned lds_base, int tid) {
#pragma unroll
    for (int i = 0; i < CHNK_; ++i) {
        const int chunk = i * 256 + tid;              // 16B chunk index
        const unsigned lds = lds_base +
            (unsigned)((chunk >> 5) * (PSTR_ * 4) + (chunk & 31) * 16);
        const float* gp = g + ((size_t)chunk << 2);
        asm volatile("global_load_async_to_lds_b128 %0, %1, off"
                     :: "v"(lds), "v"(gp) : "memory");
    }
}

__global__ __launch_bounds__(256)
void gram_kernel(const float* __restrict__ xs, float* __restrict__ G) {
    __shared__ float panel[2][NB_ * PSTR_];           // 2 x 21120 B
    const int tid = threadIdx.x;
    const int wave = tid >> 5, lane = tid & 31;
    const int hh = lane >> 4, l16 = lane & 15;
    const int t = blockIdx.y;
    const int p0 = blockIdx.x * 16;
    const int q0 = wave * 16;
    const float* base = xs + (size_t)t * N_ * P_;

    const unsigned lds0 = (unsigned)(size_t)(&panel[0][0]);
    const unsigned lds1 = (unsigned)(size_t)(&panel[1][0]);

    // prologue: stage 0 -> buffer 0
    gram_stage_issue(base, lds0, tid);

    v8f acc0 = {}, acc1 = {};
    for (int s = 0; s < NSTG_; ++s) {
        const unsigned nxt = (s & 1) ? lds0 : lds1;
        if (s + 1 < NSTG_) {
            gram_stage_issue(base + (size_t)(s + 1) * NB_ * P_, nxt, tid);
            asm volatile("s_wait_asynccnt 0x5" ::: "memory");  // stage s done
        } else {
            asm volatile("s_wait_asynccnt 0x0" ::: "memory");
        }
        __syncthreads();                       // panel visible to all 8 waves

        const float* pan = (s & 1) ? &panel[1][0] : &panel[0][0];
#pragma unroll
        for (int kk = 0; kk < NB_; kk += 8) {
            {
                const int r = kk + 2 * hh;
                v2f a, b;
                a.x = pan[r * PSTR_ + p0 + l16];
                a.y = pan[(r + 1) * PSTR_ + p0 + l16];
                b.x = pan[r * PSTR_ + q0 + l16];
                b.y = pan[(r + 1) * PSTR_ + q0 + l16];
                acc0 = __builtin_amdgcn_wmma_f32_16x16x4_f32(
                    false, a, false, b, (short)0, acc0, false, false);
            }
            {
                const int r = kk + 4 + 2 * hh;
                v2f a, b;
                a.x = pan[r * PSTR_ + p0 + l16];
                a.y = pan[(r + 1) * PSTR_ + p0 + l16];
                b.x = pan[r * PSTR_ + q0 + l16];
                b.y = pan[(r + 1) * PSTR_ + q0 + l16];
                acc1 = __builtin_amdgcn_wmma_f32_16x16x4_f32(
                    false, a, false, b, (short)0, acc1, false, false);
            }
        }
        __syncthreads();                       // all reads done before reuse
    }
#pragma unroll
    for (int i = 0; i < 8; ++i) {
        const int p = p0 + i + 8 * hh;
        G[((size_t)t * P_ + p) * P_ + q0 + l16] = acc0[i] + acc1[i];
    }
}

// ---------------------------------------------------------------------------
// zr[t,p] = sum_n xs[t,n,p] * y[t,n]
// ---------------------------------------------------------------------------
__global__ __launch_bounds__(256)
void zr_kernel(const float* __restrict__ xs, const float* __restrict__ y,
               float* __restrict__ zr) {
    const int t = blockIdx.x, tid = threadIdx.x;
    const int p = tid & (P_ - 1), slice = tid >> 7;   // 2 slices over n
    float s = 0.f;
    for (int n = slice; n < N_; n += 2)
        s += xs[((size_t)t * N_ + n) * P_ + p] * y[(size_t)t * N_ + n];
    __shared__ float red[256];
    red[tid] = s;
    __syncthreads();
    if (slice == 0) zr[t * P_ + p] = red[p] + red[P_ + p];
}

// ---------------------------------------------------------------------------
// Per-t SPD solve: G x = zr via in-LDS Cholesky + tri-solves, then
// factor[t] = Wf x + bf. (pinv(G)==inv(G): G full-rank for N>>P.)
// ---------------------------------------------------------------------------
__global__ __launch_bounds__(128)
void solve_kernel(const float* __restrict__ G, const float* __restrict__ zr,
                  const float* __restrict__ Wf, const float* __restrict__ bf,
                  float* __restrict__ factor) {
    __shared__ float A[P_ * (P_ + 1)];     // padded rows
    __shared__ float yv[P_];
    const int t = blockIdx.x, i = threadIdx.x;

    for (int idx = i; idx < P_ * P_; idx += 128)
        A[(idx / P_) * (P_ + 1) + (idx % P_)] = G[(size_t)t * P_ * P_ + idx];
    yv[i] = zr[t * P_ + i];
    __syncthreads();

    // Cholesky (lower), right-looking, thread i owns row i
    for (int k = 0; k < P_; ++k) {
        if (i == k) A[k * (P_ + 1) + k] = sqrtf(A[k * (P_ + 1) + k]);
        __syncthreads();
        const float dk = A[k * (P_ + 1) + k];
        if (i > k) A[i * (P_ + 1) + k] /= dk;
        __syncthreads();
        if (i > k) {
            const float lik = A[i * (P_ + 1) + k];
            for (int j = k + 1; j <= i; ++j)
                A[i * (P_ + 1) + j] -= lik * A[j * (P_ + 1) + k];
        }
        __syncthreads();
    }
    // forward: L y = b
    for (int k = 0; k < P_; ++k) {
        if (i == k) yv[k] /= A[k * (P_ + 1) + k];
        __syncthreads();
        const float yk = yv[k];
        if (i > k) yv[i] -= A[i * (P_ + 1) + k] * yk;
        __syncthreads();
    }
    // backward: L^T x = y
    for (int k = P_ - 1; k >= 0; --k) {
        if (i == k) yv[k] /= A[k * (P_ + 1) + k];
        __syncthreads();
        const float xk = yv[k];
        if (i < k) yv[i] -= A[k * (P_ + 1) + i] * xk;
        __syncthreads();
    }
    if (i < K_) {
        float acc = bf[i];
        for (int p = 0; p < P_; ++p) acc += Wf[i * P_ + p] * yv[p];
        factor[t * K_ + i] = acc;
    }
}

// ---------------------------------------------------------------------------
// Head: beta = relu(bn(H1pre)) @ W2^T + b2 ; r_pred = beta . factor ;
// per-block partial SSE (deterministic two-stage reduction).
// ---------------------------------------------------------------------------
__global__ __launch_bounds__(256)
void head_kernel(const float* __restrict__ H1pre, const float* __restrict__ sc1,
                 const float* __restrict__ sh1, const float* __restrict__ W2,
                 const float* __restrict__ b2, const float* __restrict__ factor,
                 const float* __restrict__ y, float* __restrict__ out,
                 float* __restrict__ psum) {
    const int id = blockIdx.x * 256 + threadIdx.x;
    float e2 = 0.f;
    if (id < T_ * N_) {
        const int t = id / N_;
        float h[H1_];
        const float* hp = H1pre + (size_t)id * H1_;
        const float* sc = sc1 + t * H1_;
        const float* sh = sh1 + t * H1_;
#pragma unroll
        for (int j = 0; j < H1_; ++j) h[j] = fmaxf(hp[j] * sc[j] + sh[j], 0.f);
        float r = 0.f;
#pragma unroll
        for (int k = 0; k < K_; ++k) {
            float bk = b2[k];
#pragma unroll
            for (int j = 0; j < H1_; ++j) bk += W2[k * H1_ + j] * h[j];
            r += bk * factor[t * K_ + k];
        }
        out[1 + id] = r;
        const float e = r - y[id];
        e2 = e * e;
    }
    __shared__ float red[256];
    red[threadIdx.x] = e2;
    __syncthreads();
    for (int s = 128; s > 0; s >>= 1) {
        if (threadIdx.x < s) red[threadIdx.x] += red[threadIdx.x + s];
        __syncthreads();
    }
    if (threadIdx.x == 0) psum[blockIdx.x] = red[0];
}

__global__ __launch_bounds__(256)
void loss_kernel(const float* __restrict__ psum, int nblocks, float* __restrict__ out) {
    __shared__ float red[256];
    float s = 0.f;
    for (int i = threadIdx.x; i < nblocks; i += 256) s += psum[i];
    red[threadIdx.x] = s;
    __syncthreads();
    for (int k = 128; k > 0; k >>= 1) {
        if (threadIdx.x < k) red[threadIdx.x] += red[threadIdx.x + k];
        __syncthreads();
    }
    if (threadIdx.x == 0) out[0] = red[0] * (1.f / ((float)T_ * (float)N_));
}

// ---------------------------------------------------------------------------
extern "C" void kernel_launch(void* const* d_in, const int* in_sizes, int n_in,
                              void* d_out, int out_size, void* d_ws, size_t ws_size,
                              hipStream_t stream) {
    const float* xs  = (const float*)d_in[0];
    const float* y   = (const float*)d_in[1];
    const float* W0  = (const float*)d_in[2];
    const float* b0  = (const float*)d_in[3];
    const float* g0  = (const float*)d_in[4];
    const float* be0 = (const float*)d_in[5];
    const float* W1  = (const float*)d_in[6];
    const float* b1  = (const float*)d_in[7];
    const float* g1  = (const float*)d_in[8];
    const float* be1 = (const float*)d_in[9];
    const float* W2  = (const float*)d_in[10];
    const float* b2  = (const float*)d_in[11];
    const float* Wf  = (const float*)d_in[12];
    const float* bf  = (const float*)d_in[13];
    float* out = (float*)d_out;
    float* ws  = (float*)d_ws;

    size_t o = 0;
    float* H0pre = ws + o; o += (size_t)T_ * N_ * H0_;
    float* H1pre = ws + o; o += (size_t)T_ * N_ * H1_;
    float* sc0   = ws + o; o += (size_t)T_ * H0_;
    float* sh0   = ws + o; o += (size_t)T_ * H0_;
    float* sc1   = ws + o; o += (size_t)T_ * H1_;
    float* sh1   = ws + o; o += (size_t)T_ * H1_;
    float* Gm    = ws + o; o += (size_t)T_ * P_ * P_;
    float* zrv   = ws + o; o += (size_t)T_ * P_;
    float* fac   = ws + o; o += (size_t)T_ * K_;
    const int nblk = (T_ * N_ + 255) / 256;
    float* psum  = ws + o; o += (size_t)nblk;

    gemm1_kernel<<<dim3((N_ / 16 + 7) / 8, T_), dim3(256), 0, stream>>>(xs, W0, b0, H0pre);
    bnstats_kernel<H0_, 256 / H0_><<<dim3(T_), dim3(256), 0, stream>>>(H0pre, g0, be0, sc0, sh0);
    gemm2_kernel<<<dim3((N_ / 16 + 7) / 8, T_), dim3(256), 0, stream>>>(H0pre, sc0, sh0, W1, b1, H1pre);
    bnstats_kernel<H1_, 256 / H1_><<<dim3(T_), dim3(256), 0, stream>>>(H1pre, g1, be1, sc1, sh1);
    gram_kernel<<<dim3(P_ / 16, T_), dim3(256), 0, stream>>>(xs, Gm);
    zr_kernel<<<dim3(T_), dim3(256), 0, stream>>>(xs, y, zrv);
    solve_kernel<<<dim3(T_), dim3(128), 0, stream>>>(Gm, zrv, Wf, bf, fac);
    head_kernel<<<dim3(nblk), dim3(256), 0, stream>>>(H1pre, sc1, sh1, W2, b2, fac, y, out, psum);
    loss_kernel<<<dim3(1), dim3(256), 0, stream>>>(psum, nblk, out);
}